// LEnsemble_55963423867090
// MI455X (gfx1250) — compile-verified
//
#include <hip/hip_runtime.h>
#include <math.h>

typedef float v2f __attribute__((ext_vector_type(2)));
typedef float v8f __attribute__((ext_vector_type(8)));

#define NDIM  512
#define NJOBS 257      // 256 samples + 1 normalizer (L + I)
#define NT16  32       // 512/16 tiles per dim
#define NTRI  (NT16 * (NT16 + 1) / 2)   // 528 lower-tri tiles
#define PST   18       // LDS panel row stride (even -> 8B-aligned v2f loads)

// ---------------- Kernel 0: compact selected indices per sample ----------------
__global__ void k_compact(const int* __restrict__ x, int* __restrict__ idx,
                          int* __restrict__ scnt)
{
    const int j = blockIdx.x;
    if (threadIdx.x != 0) return;
    int* id = idx + (size_t)j * NDIM;
    if (j == NJOBS - 1) {                       // normalizer job: full index set
        for (int a = 0; a < NDIM; ++a) id[a] = a;
        scnt[j] = NDIM;
        return;
    }
    const int* xr = x + (size_t)j * NDIM;
    int c = 0;
    for (int a = 0; a < NDIM; ++a) if (xr[a] == 1) id[c++] = a;
    scnt[j] = c;
    for (int a = c; a < NDIM; ++a) id[a] = a;   // safe pad rows (overwritten later)
}

// ---------------- Kernel 1: L = B * B^T via f32 WMMA ---------------------------
// One wave per 16x16 tile, V_WMMA_F32_16X16X4_F32, K = 512 (128 steps).
__global__ void k_lgemm(const float* __restrict__ B, float* __restrict__ L)
{
    const int i0   = blockIdx.y * 16;
    const int j0   = blockIdx.x * 16;
    const int lane = threadIdx.x;     // 0..31
    const int m    = lane & 15;
    const int half = lane >> 4;       // 0/1
    const int col  = j0 + m;
    const int rb   = i0 + half * 8;   // C/D layout: VGPR e -> row rb+e

    const float* arow = B + (size_t)(i0 + m) * NDIM;   // A rows (M side)
    const float* brow = B + (size_t)(j0 + m) * NDIM;   // B^T rows (N side)
    const int koff = half * 2;        // lanes 16-31 hold K+2,K+3

    v8f c = {};
    #pragma unroll 4
    for (int k = 0; k < NDIM; k += 4) {
        v2f a = *(const v2f*)(arow + k + koff);
        v2f b = *(const v2f*)(brow + k + koff);
        c = __builtin_amdgcn_wmma_f32_16x16x4_f32(false, a, false, b,
                                                  (short)0, c, false, false);
    }
    #pragma unroll
    for (int e = 0; e < 8; ++e)
        L[(size_t)(rb + e) * NDIM + col] = c[e];
}

// ---------------- Kernel 2: gather compacted job matrices from L ---------------
// Only lower block-triangle tiles (bi >= bj) are materialized: the Cholesky
// touches exactly those. C_j[a][b] = L[idx_a][idx_b] (+I for job 256),
// identity in the pad region a>=s or b>=s.
__global__ void k_gather(const float* __restrict__ L, const int* __restrict__ idx,
                         const int* __restrict__ scnt, float* __restrict__ slots,
                         int baseJob)
{
    const int t = blockIdx.x;                 // 0..NTRI-1 -> (bi,bj), bi>=bj
    int bi = 0;
    while ((bi + 1) * (bi + 2) / 2 <= t) ++bi;
    const int bj = t - bi * (bi + 1) / 2;

    const int job = baseJob + blockIdx.y;
    const int s   = scnt[job];
    float* C = slots + (size_t)blockIdx.y * NDIM * NDIM;
    const int* id = idx + (size_t)job * NDIM;
    const bool addI = (job == NJOBS - 1);

    const int r   = threadIdx.x >> 4;         // 256 threads = one 16x16 tile
    const int cc  = threadIdx.x & 15;
    const int row = bi * 16 + r;
    const int col = bj * 16 + cc;

    float v;
    if (row < s && col < s) {
        v = L[(size_t)id[row] * NDIM + id[col]];
        if (addI && row == col) v += 1.0f;
    } else {
        v = (row == col) ? 1.0f : 0.0f;       // identity padding
    }
    C[(size_t)row * NDIM + col] = v;
}

// ---------------- Kernel 3: blocked Cholesky logdet (nb=16), WMMA SYRK ---------
// One 256-thread workgroup (8 waves) per matrix. Trailing update
// C -= P * P^T via V_WMMA_F32_16X16X4_F32 (A negated at load; f32 WMMA NEG is
// CNeg-only). Factors of finished block columns are never written back.
__global__ void k_chol(float* __restrict__ slots, const int* __restrict__ scnt,
                       float* __restrict__ logd, int baseJob)
{
    __shared__ float Dl[16 * 17];     // diagonal block (lower factor in place)
    __shared__ float P [496 * PST];   // solved panel rows (max 512-16 rows)

    const int job = baseJob + blockIdx.x;
    float* C = slots + (size_t)blockIdx.x * NDIM * NDIM;
    const int s   = scnt[job];
    const int M16 = (s + 15) >> 4;    // active blocks (pad rows -> identity, log 0)
    const int M   = M16 << 4;

    const int tid  = threadIdx.x;     // 0..255
    const int lane = tid & 31;
    const int wave = tid >> 5;
    float lacc = 0.0f;                // meaningful on tid 0 only

    for (int p = 0; p < M16; ++p) {
        const int k0 = p << 4;
        // (a) stage diagonal block into LDS
        {
            const int r = tid >> 4, cc = tid & 15;
            Dl[r * 17 + cc] = C[(size_t)(k0 + r) * NDIM + (k0 + cc)];
        }
        __syncthreads();
        // serial 16x16 Cholesky (trivial cost) + logdet accumulation
        if (tid == 0) {
            for (int r = 0; r < 16; ++r) {
                float d = sqrtf(Dl[r * 17 + r]);
                Dl[r * 17 + r] = d;
                lacc += logf(d);
                const float inv = 1.0f / d;
                for (int i = r + 1; i < 16; ++i) Dl[i * 17 + r] *= inv;
                for (int j2 = r + 1; j2 < 16; ++j2) {
                    const float f = Dl[j2 * 17 + r];
                    for (int i = j2; i < 16; ++i)
                        Dl[i * 17 + j2] -= Dl[i * 17 + r] * f;
                }
            }
        }
        __syncthreads();

        const int rs = k0 + 16;       // trailing-region start
        // (b) panel solve: X * Lpp^T = A  -> P (forward substitution per row)
        for (int i = rs + tid; i < M; i += 256) {
            const float* arow = C + (size_t)i * NDIM + k0;
            float xr[16];
            #pragma unroll
            for (int cc = 0; cc < 16; ++cc) {
                float v = arow[cc];
                for (int r = 0; r < cc; ++r) v -= xr[r] * Dl[cc * 17 + r];
                xr[cc] = v / Dl[cc * 17 + cc];
            }
            float* prow = P + (size_t)(i - rs) * PST;
            #pragma unroll
            for (int cc = 0; cc < 16; ++cc) prow[cc] = xr[cc];
        }
        __syncthreads();

        // (c) trailing SYRK: lower-triangle tiles, round-robin over 8 waves
        {
            const int m    = lane & 15;
            const int half = lane >> 4;
            int t = 0;
            for (int bi = p + 1; bi < M16; ++bi) {
                for (int bj = p + 1; bj <= bi; ++bj, ++t) {
                    if ((t & 7) != wave) continue;      // wave-uniform
                    const int row0 = bi << 4, col0 = bj << 4;
                    const int colg = col0 + m;
                    const int rb   = row0 + half * 8;
                    v8f c;
                    #pragma unroll
                    for (int e = 0; e < 8; ++e)
                        c[e] = C[(size_t)(rb + e) * NDIM + colg];
                    const float* pa = P + (size_t)(row0 + m - rs) * PST + half * 2;
                    const float* pb = P + (size_t)(col0 + m - rs) * PST + half * 2;
                    #pragma unroll
                    for (int kb = 0; kb < 16; kb += 4) {   // rank-16 via 4x K=4
                        v2f a; a.x = -pa[kb]; a.y = -pa[kb + 1];  // C - P*P^T
                        v2f b = *(const v2f*)(pb + kb);
                        c = __builtin_amdgcn_wmma_f32_16x16x4_f32(false, a, false, b,
                                                                  (short)0, c, false, false);
                    }
                    #pragma unroll
                    for (int e = 0; e < 8; ++e)
                        C[(size_t)(rb + e) * NDIM + colg] = c[e];
                }
            }
        }
        __syncthreads();
    }
    if (tid == 0) logd[job] = 2.0f * lacc;
}

// ---------------- Kernel 4: out[j] = logdet_j - logdet_norm --------------------
__global__ void k_fin(const float* __restrict__ logd, float* __restrict__ out)
{
    const int t = threadIdx.x;
    if (t < 256) out[t] = logd[t] - logd[NJOBS - 1];
}

extern "C" void kernel_launch(void* const* d_in, const int* in_sizes, int n_in,
                              void* d_out, int out_size, void* d_ws, size_t ws_size,
                              hipStream_t stream)
{
    const float* B = (const float*)d_in[0];
    const int*   x = (const int*)d_in[1];
    float* out = (float*)d_out;

    // Workspace: [idx 257*512 i32][scnt 257][logd 257][L 512*512 f32][slots ...]
    char* ws = (char*)d_ws;
    int*   idx  = (int*)ws;
    int*   scnt = idx + (size_t)NJOBS * NDIM;
    float* logd = (float*)(scnt + NJOBS);
    size_t off = (size_t)NJOBS * NDIM * 4 + (size_t)NJOBS * 4 + (size_t)NJOBS * 4;
    off = (off + 255) & ~(size_t)255;
    float* L = (float*)(ws + off);
    off += (size_t)NDIM * NDIM * sizeof(float);
    off = (off + 255) & ~(size_t)255;
    float* slots = (float*)(ws + off);

    const size_t slot_bytes = (size_t)NDIM * NDIM * sizeof(float);
    size_t avail = (ws_size > off) ? (ws_size - off) : 0;
    int nslots = (int)(avail / slot_bytes);
    if (nslots < 1)     nslots = 1;       // best effort if scratch is tiny
    if (nslots > NJOBS) nslots = NJOBS;

    k_compact<<<NJOBS, 32, 0, stream>>>(x, idx, scnt);
    k_lgemm<<<dim3(NT16, NT16), 32, 0, stream>>>(B, L);
    for (int baseJob = 0; baseJob < NJOBS; baseJob += nslots) {
        int cnt = NJOBS - baseJob;
        if (cnt > nslots) cnt = nslots;
        k_gather<<<dim3(NTRI, cnt), 256, 0, stream>>>(L, idx, scnt, slots, baseJob);
        k_chol<<<cnt, 256, 0, stream>>>(slots, scnt, logd, baseJob);
    }
    k_fin<<<1, 256, 0, stream>>>(logd, out);
}